// DiceBCE_OHNMLoss_5119601017433
// MI455X (gfx1250) — compile-verified
//
#include <hip/hip_runtime.h>
#include <stdint.h>

// ---------------- workspace layout (uint32 words) ----------------
#define HIST0 4096   // key bits [31:20]
#define HIST1 4096   // key bits [19:8]
#define HIST2 256    // key bits [7:0]

#define WS_H0_CNT   0
#define WS_H0_LOSS  (WS_H0_CNT + HIST0)
#define WS_H0_SIG   (WS_H0_LOSS + HIST0)
#define WS_H1_CNT   (WS_H0_SIG + HIST0)
#define WS_H1_LOSS  (WS_H1_CNT + HIST1)
#define WS_H1_SIG   (WS_H1_LOSS + HIST1)
#define WS_H2_CNT   (WS_H1_SIG + HIST1)
#define WS_H2_LOSS  (WS_H2_CNT + HIST2)
#define WS_H2_SIG   (WS_H2_LOSS + HIST2)
#define WS_STATS    (WS_H2_SIG + HIST2)
#define ST_NPOS   0
#define ST_PLOSS  1
#define ST_PSIG   2
#define ST_NHNS   3
#define ST_BIN0   4
#define ST_REM0   5
#define ST_BIN1   6
#define ST_REM1   7
#define ST_ACCL   8
#define ST_ACCS   9
#define WS_TOTAL  (WS_STATS + 16)   // 25360 words = ~99 KB

#define SENTINEL 0xFFFFFFFFu

// order-preserving float->uint key
__device__ __forceinline__ uint32_t f2key(float p) {
  uint32_t u = __float_as_uint(p);
  return (u & 0x80000000u) ? ~u : (u | 0x80000000u);
}
__device__ __forceinline__ float key2f(uint32_t k) {
  uint32_t u = (k & 0x80000000u) ? (k & 0x7FFFFFFFu) : ~k;
  return __uint_as_float(u);
}
// logaddexp(0,p) = softplus(p), numerically stable (matches reference)
__device__ __forceinline__ float softplusf(float p) {
  return log1pf(expf(-fabsf(p))) + fmaxf(p, 0.f);
}
__device__ __forceinline__ float sigmoidf(float p) {
  return 1.f / (1.f + expf(-p));
}

// ---------------- kernel 0: zero workspace ----------------
__global__ void k_zero(uint32_t* __restrict__ ws, int nwords) {
  int i = blockIdx.x * blockDim.x + threadIdx.x;
  if (i < nwords) ws[i] = 0u;
}

// per-element processing for pass1
__device__ __forceinline__ void proc1(float p, int tv,
                                      uint32_t* hcnt, float* hloss, float* hsig,
                                      uint32_t& l_np, float& l_ploss, float& l_psig) {
  float sp = softplusf(p);
  float sg = sigmoidf(p);
  if (tv != 0) {
    l_np += 1u; l_ploss += sp - p; l_psig += sg;   // BCE with t=1: softplus(p)-p
  } else {
    uint32_t b = f2key(p) >> 20;
    atomicAdd(&hcnt[b], 1u);
    atomicAdd(&hloss[b], sp);
    atomicAdd(&hsig[b], sg);
  }
}

// ---------------- pass1: stats + level-0 histogram ----------------
// Streams preds+targs through LDS via double-buffered async global->LDS copies.
#define T1TILE 512
__global__ __launch_bounds__(256) void k_pass1(const float* __restrict__ preds,
                                               const int* __restrict__ targs,
                                               int n, uint32_t* __restrict__ ws) {
  __shared__ __align__(16) float pbuf[2][T1TILE];
  __shared__ __align__(16) int   tbuf[2][T1TILE];
  __shared__ uint32_t hcnt[HIST0];
  __shared__ float    hloss[HIST0];
  __shared__ float    hsig[HIST0];
  const int t = threadIdx.x;
  for (int i = t; i < HIST0; i += 256) { hcnt[i] = 0u; hloss[i] = 0.f; hsig[i] = 0.f; }
  __syncthreads();

  uint32_t l_np = 0u; float l_ploss = 0.f, l_psig = 0.f;

  const int ntiles = n / T1TILE;
  int tile = blockIdx.x;
  int buf = 0;
  if (tile < ntiles) {
    const float* gp = preds + (size_t)tile * T1TILE + t * 2;
    const int*   gt = targs + (size_t)tile * T1TILE + t * 2;
    uint32_t lp = (uint32_t)(uint64_t)&pbuf[0][t * 2];
    uint32_t lt = (uint32_t)(uint64_t)&tbuf[0][t * 2];
    asm volatile("global_load_async_to_lds_b64 %0, %2, off\n\t"
                 "global_load_async_to_lds_b64 %1, %3, off"
                 :: "v"(lp), "v"(lt), "v"(gp), "v"(gt) : "memory");
  }
  while (tile < ntiles) {
    int nxt = tile + (int)gridDim.x;
    if (nxt < ntiles) {
      const float* gp = preds + (size_t)nxt * T1TILE + t * 2;
      const int*   gt = targs + (size_t)nxt * T1TILE + t * 2;
      uint32_t lp = (uint32_t)(uint64_t)&pbuf[buf ^ 1][t * 2];
      uint32_t lt = (uint32_t)(uint64_t)&tbuf[buf ^ 1][t * 2];
      asm volatile("global_load_async_to_lds_b64 %0, %2, off\n\t"
                   "global_load_async_to_lds_b64 %1, %3, off"
                   :: "v"(lp), "v"(lt), "v"(gp), "v"(gt) : "memory");
      asm volatile("s_wait_asynccnt 0x2" ::: "memory");   // current tile landed
    } else {
      asm volatile("s_wait_asynccnt 0x0" ::: "memory");
    }
    // each thread consumes exactly the 2 elements it loaded (no cross-lane LDS)
    #pragma unroll
    for (int j = 0; j < 2; ++j) {
      proc1(pbuf[buf][t * 2 + j], tbuf[buf][t * 2 + j],
            hcnt, hloss, hsig, l_np, l_ploss, l_psig);
    }
    buf ^= 1;
    tile = nxt;
  }
  // tail (n % T1TILE) handled by block 0 with plain loads
  if (blockIdx.x == 0) {
    for (int i = ntiles * T1TILE + t; i < n; i += 256)
      proc1(preds[i], targs[i], hcnt, hloss, hsig, l_np, l_ploss, l_psig);
  }
  __syncthreads();

  uint32_t* g_cnt  = ws + WS_H0_CNT;
  float*    g_loss = (float*)(ws + WS_H0_LOSS);
  float*    g_sig  = (float*)(ws + WS_H0_SIG);
  for (int i = t; i < HIST0; i += 256) {
    uint32_t c = hcnt[i];
    if (c) { atomicAdd(&g_cnt[i], c); atomicAdd(&g_loss[i], hloss[i]); atomicAdd(&g_sig[i], hsig[i]); }
  }
  uint32_t* stats = ws + WS_STATS;
  if (l_np) {
    atomicAdd(&stats[ST_NPOS], l_np);
    atomicAdd((float*)&stats[ST_PLOSS], l_ploss);
    atomicAdd((float*)&stats[ST_PSIG],  l_psig);
  }
}

// ---------------- select0: n_hns, level-0 bin, sums above ----------------
__global__ void k_select0(int n, uint32_t* __restrict__ ws) {
  __shared__ uint32_t cnt[HIST0];
  __shared__ float lo[HIST0], si[HIST0];
  float* wsf = (float*)ws;
  for (int i = threadIdx.x; i < HIST0; i += blockDim.x) {
    cnt[i] = ws[WS_H0_CNT + i];
    lo[i]  = wsf[WS_H0_LOSS + i];
    si[i]  = wsf[WS_H0_SIG + i];
  }
  __syncthreads();
  if (threadIdx.x == 0) {
    uint32_t* stats = ws + WS_STATS; float* statsf = (float*)stats;
    uint32_t n_pos = stats[ST_NPOS];
    uint32_t n_neg = (uint32_t)n - n_pos;
    uint32_t n_hns;
    if (n_pos == 0u) n_hns = (uint32_t)(0.1 * (double)n_neg);
    else {
      unsigned long long h = (unsigned long long)n_pos * 30ull;
      n_hns = (h < (unsigned long long)n_neg) ? (uint32_t)h : n_neg;
    }
    stats[ST_NHNS] = n_hns;
    uint32_t bin0 = SENTINEL, rem0 = 0u;
    float accl = 0.f, accs = 0.f;
    if (n_hns > 0u) {
      unsigned long long cum = 0ull;
      for (int i = HIST0 - 1; i >= 0; --i) {
        uint32_t c = cnt[i];
        if (cum + c >= (unsigned long long)n_hns) { bin0 = (uint32_t)i; rem0 = n_hns - (uint32_t)cum; break; }
        cum += c; accl += lo[i]; accs += si[i];
      }
    }
    stats[ST_BIN0] = bin0; stats[ST_REM0] = rem0;
    statsf[ST_ACCL] = accl; statsf[ST_ACCS] = accs;
  }
}

// ---------------- pass2: level-1 histogram of candidates ----------------
__global__ __launch_bounds__(256) void k_pass2(const float* __restrict__ preds,
                                               const int* __restrict__ targs,
                                               int n, uint32_t* __restrict__ ws) {
  __shared__ uint32_t hcnt[HIST1];
  __shared__ float hloss[HIST1], hsig[HIST1];
  const uint32_t bin0 = ws[WS_STATS + ST_BIN0];
  for (int i = threadIdx.x; i < HIST1; i += 256) { hcnt[i] = 0u; hloss[i] = 0.f; hsig[i] = 0.f; }
  __syncthreads();
  if (bin0 != SENTINEL) {
    int idx = blockIdx.x * blockDim.x + threadIdx.x;
    int stride = gridDim.x * blockDim.x;
    int n4 = n >> 2;
    const float4* p4 = (const float4*)preds;
    const int4*   t4 = (const int4*)targs;
    for (int i = idx; i < n4; i += stride) {
      float4 pv = p4[i]; int4 tv = t4[i];
      float pj[4] = {pv.x, pv.y, pv.z, pv.w};
      int   tj[4] = {tv.x, tv.y, tv.z, tv.w};
      #pragma unroll
      for (int j = 0; j < 4; ++j) {
        if (tj[j] == 0) {
          uint32_t key = f2key(pj[j]);
          if ((key >> 20) == bin0) {
            uint32_t b = (key >> 8) & 0xFFFu;
            atomicAdd(&hcnt[b], 1u);
            atomicAdd(&hloss[b], softplusf(pj[j]));
            atomicAdd(&hsig[b],  sigmoidf(pj[j]));
          }
        }
      }
    }
    if (blockIdx.x == 0) {
      for (int i = (n4 << 2) + threadIdx.x; i < n; i += blockDim.x) {
        if (targs[i] == 0) {
          float p = preds[i]; uint32_t key = f2key(p);
          if ((key >> 20) == bin0) {
            uint32_t b = (key >> 8) & 0xFFFu;
            atomicAdd(&hcnt[b], 1u);
            atomicAdd(&hloss[b], softplusf(p));
            atomicAdd(&hsig[b],  sigmoidf(p));
          }
        }
      }
    }
  }
  __syncthreads();
  uint32_t* g_cnt  = ws + WS_H1_CNT;
  float*    g_loss = (float*)(ws + WS_H1_LOSS);
  float*    g_sig  = (float*)(ws + WS_H1_SIG);
  for (int i = threadIdx.x; i < HIST1; i += 256) {
    uint32_t c = hcnt[i];
    if (c) { atomicAdd(&g_cnt[i], c); atomicAdd(&g_loss[i], hloss[i]); atomicAdd(&g_sig[i], hsig[i]); }
  }
}

// ---------------- select1 ----------------
__global__ void k_select1(uint32_t* __restrict__ ws) {
  __shared__ uint32_t cnt[HIST1];
  __shared__ float lo[HIST1], si[HIST1];
  float* wsf = (float*)ws;
  for (int i = threadIdx.x; i < HIST1; i += blockDim.x) {
    cnt[i] = ws[WS_H1_CNT + i];
    lo[i]  = wsf[WS_H1_LOSS + i];
    si[i]  = wsf[WS_H1_SIG + i];
  }
  __syncthreads();
  if (threadIdx.x == 0) {
    uint32_t* stats = ws + WS_STATS; float* statsf = (float*)stats;
    uint32_t rem0 = stats[ST_REM0];
    uint32_t bin1 = SENTINEL, rem1 = 0u;
    float accl = statsf[ST_ACCL], accs = statsf[ST_ACCS];
    if (rem0 > 0u && stats[ST_BIN0] != SENTINEL) {
      unsigned long long cum = 0ull;
      for (int i = HIST1 - 1; i >= 0; --i) {
        uint32_t c = cnt[i];
        if (cum + c >= (unsigned long long)rem0) { bin1 = (uint32_t)i; rem1 = rem0 - (uint32_t)cum; break; }
        cum += c; accl += lo[i]; accs += si[i];
      }
    }
    stats[ST_BIN1] = bin1; stats[ST_REM1] = rem1;
    statsf[ST_ACCL] = accl; statsf[ST_ACCS] = accs;
  }
}

// ---------------- pass3: level-2 histogram ----------------
__global__ __launch_bounds__(256) void k_pass3(const float* __restrict__ preds,
                                               const int* __restrict__ targs,
                                               int n, uint32_t* __restrict__ ws) {
  __shared__ uint32_t hcnt[HIST2];
  __shared__ float hloss[HIST2], hsig[HIST2];
  const uint32_t bin0 = ws[WS_STATS + ST_BIN0];
  const uint32_t bin1 = ws[WS_STATS + ST_BIN1];
  for (int i = threadIdx.x; i < HIST2; i += 256) { hcnt[i] = 0u; hloss[i] = 0.f; hsig[i] = 0.f; }
  __syncthreads();
  if (bin0 != SENTINEL && bin1 != SENTINEL) {
    const uint32_t top24 = (bin0 << 12) | bin1;
    int idx = blockIdx.x * blockDim.x + threadIdx.x;
    int stride = gridDim.x * blockDim.x;
    int n4 = n >> 2;
    const float4* p4 = (const float4*)preds;
    const int4*   t4 = (const int4*)targs;
    for (int i = idx; i < n4; i += stride) {
      float4 pv = p4[i]; int4 tv = t4[i];
      float pj[4] = {pv.x, pv.y, pv.z, pv.w};
      int   tj[4] = {tv.x, tv.y, tv.z, tv.w};
      #pragma unroll
      for (int j = 0; j < 4; ++j) {
        if (tj[j] == 0) {
          uint32_t key = f2key(pj[j]);
          if ((key >> 8) == top24) {
            uint32_t b = key & 0xFFu;
            atomicAdd(&hcnt[b], 1u);
            atomicAdd(&hloss[b], softplusf(pj[j]));
            atomicAdd(&hsig[b],  sigmoidf(pj[j]));
          }
        }
      }
    }
    if (blockIdx.x == 0) {
      for (int i = (n4 << 2) + threadIdx.x; i < n; i += blockDim.x) {
        if (targs[i] == 0) {
          float p = preds[i]; uint32_t key = f2key(p);
          if ((key >> 8) == top24) {
            uint32_t b = key & 0xFFu;
            atomicAdd(&hcnt[b], 1u);
            atomicAdd(&hloss[b], softplusf(p));
            atomicAdd(&hsig[b],  sigmoidf(p));
          }
        }
      }
    }
  }
  __syncthreads();
  uint32_t* g_cnt  = ws + WS_H2_CNT;
  float*    g_loss = (float*)(ws + WS_H2_LOSS);
  float*    g_sig  = (float*)(ws + WS_H2_SIG);
  for (int i = threadIdx.x; i < HIST2; i += 256) {
    uint32_t c = hcnt[i];
    if (c) { atomicAdd(&g_cnt[i], c); atomicAdd(&g_loss[i], hloss[i]); atomicAdd(&g_sig[i], hsig[i]); }
  }
}

// ---------------- select2 + finalize ----------------
__global__ void k_select2(uint32_t* __restrict__ ws, float* __restrict__ out) {
  __shared__ uint32_t cnt[HIST2];
  __shared__ float lo[HIST2], si[HIST2];
  float* wsf = (float*)ws;
  for (int i = threadIdx.x; i < HIST2; i += blockDim.x) {
    cnt[i] = ws[WS_H2_CNT + i];
    lo[i]  = wsf[WS_H2_LOSS + i];
    si[i]  = wsf[WS_H2_SIG + i];
  }
  __syncthreads();
  if (threadIdx.x == 0) {
    uint32_t* stats = ws + WS_STATS; float* statsf = (float*)stats;
    uint32_t rem1 = stats[ST_REM1];
    uint32_t bin0 = stats[ST_BIN0], bin1 = stats[ST_BIN1];
    float accl = statsf[ST_ACCL], accs = statsf[ST_ACCS];
    if (rem1 > 0u && bin0 != SENTINEL && bin1 != SENTINEL) {
      unsigned long long cum = 0ull;
      for (int i = HIST2 - 1; i >= 0; --i) {
        uint32_t c = cnt[i];
        if (cum + c >= (unsigned long long)rem1) {
          uint32_t r = rem1 - (uint32_t)cum;       // ties at exact threshold key
          uint32_t key = (bin0 << 20) | (bin1 << 8) | (uint32_t)i;
          float pt = key2f(key);
          accl += (float)r * softplusf(pt);
          accs += (float)r * sigmoidf(pt);
          break;
        }
        cum += c; accl += lo[i]; accs += si[i];
      }
    }
    uint32_t n_pos = stats[ST_NPOS], n_hns = stats[ST_NHNS];
    float pl = statsf[ST_PLOSS], ps = statsf[ST_PSIG];
    float totalCnt = (float)(n_hns + n_pos);
    float mean_loss = (totalCnt > 0.f) ? (accl + pl) / totalCnt : 0.f;
    float inter = ps;                               // negatives contribute 0 to ps*ts
    float denom = (accs + ps) + (float)n_pos;       // sum(sigmoid_sel) + sum(t_sel)
    float dice = 1.f - (2.f * inter + 1e-10f) / (denom + 1e-10f);
    out[0] = dice + mean_loss;
  }
}

extern "C" void kernel_launch(void* const* d_in, const int* in_sizes, int n_in,
                              void* d_out, int out_size, void* d_ws, size_t ws_size,
                              hipStream_t stream) {
  (void)n_in; (void)out_size; (void)ws_size;
  const float* preds = (const float*)d_in[0];
  const int*   targs = (const int*)d_in[1];
  int n = in_sizes[0];
  uint32_t* ws = (uint32_t*)d_ws;   // needs WS_TOTAL*4 ≈ 101 KB
  float* out = (float*)d_out;

  k_zero<<<(WS_TOTAL + 255) / 256, 256, 0, stream>>>(ws, WS_TOTAL);
  k_pass1<<<2048, 256, 0, stream>>>(preds, targs, n, ws);
  k_select0<<<1, 256, 0, stream>>>(n, ws);
  k_pass2<<<2048, 256, 0, stream>>>(preds, targs, n, ws);
  k_select1<<<1, 256, 0, stream>>>(ws);
  k_pass3<<<2048, 256, 0, stream>>>(preds, targs, n, ws);
  k_select2<<<1, 256, 0, stream>>>(ws, out);
}